// WeightsLib2DDelta_6451040878919
// MI455X (gfx1250) — compile-verified
//
#include <hip/hip_runtime.h>
#include <hip/hip_bf16.h>
#include <math.h>

#define B_    32
#define CTX_  2048
#define RANK_ 4
#define HW_   65536          // 256*256
#define OUTF_ 106            // 24*RANK + RANK*2 + 2
#define SLICE_ATT_ 53
#define PLANE_ 131072        // H*W*2 floats per (r,l,t) plane
#define EPS_  1e-12f

typedef float v2f __attribute__((ext_vector_type(2)));
typedef float v8f __attribute__((ext_vector_type(8)));

// ---------------------------------------------------------------------------
// Kernel 1: xt = x @ ctx_w.T + ctx_b  via V_WMMA_F32_16X16X4_F32
// M=32 (2 tiles) x N=112 (7 tiles) x K=2048. One wave per 16x16 tile.
// ---------------------------------------------------------------------------
__global__ __launch_bounds__(448) void k1_gemm_wmma(
    const float* __restrict__ x,      // [32][2048]
    const float* __restrict__ ctx_w,  // [106][2048]
    const float* __restrict__ ctx_b,  // [106]
    float* __restrict__ xt)           // [32][106]
{
    const int tid  = threadIdx.x;
    const int wave = tid >> 5;          // 0..13
    const int lane = tid & 31;
    const int tileM = wave / 7;         // 0..1
    const int tileN = wave % 7;         // 0..6

    const int l15   = lane & 15;
    const int khalf = (lane >> 4) * 2;  // 0 or 2

    const int mr = tileM * 16 + l15;                    // row of A (always < 32)
    int nc = tileN * 16 + l15;                          // col of B
    const int ncc = nc < OUTF_ ? nc : (OUTF_ - 1);      // clamp (branchless) for loads

    const float* __restrict__ arow = x     + mr  * CTX_ + khalf;
    const float* __restrict__ brow = ctx_w + ncc * CTX_ + khalf;

    v8f acc = {};
#pragma unroll 4
    for (int k0 = 0; k0 < CTX_; k0 += 4) {
        v2f a = *reinterpret_cast<const v2f*>(arow + k0);
        v2f b = *reinterpret_cast<const v2f*>(brow + k0);
        acc = __builtin_amdgcn_wmma_f32_16x16x4_f32(
                false, a, false, b, (short)0, acc, false, false);
    }

    // C/D layout: VGPR i -> rows tileM*16 + i (lanes 0-15) / + i + 8 (lanes 16-31)
    const int rbase = tileM * 16 + ((lane >> 4) << 3);
    const int col   = tileN * 16 + l15;
    if (col < OUTF_) {
        const float bias = ctx_b[col];
#pragma unroll
        for (int i = 0; i < 8; ++i) {
            xt[(rbase + i) * OUTF_ + col] = acc[i] + bias;
        }
    }
}

// ---------------------------------------------------------------------------
// Small math helpers
// ---------------------------------------------------------------------------
__device__ __forceinline__ float siglog_f(float v) {
    return copysignf(log1pf(fabsf(v)), v);
}

__device__ __forceinline__ unsigned rotl32(unsigned v, int r) {
    return (v << r) | (v >> (32 - r));
}

__device__ __forceinline__ void threefry2x32(unsigned k0, unsigned k1,
                                             unsigned x0, unsigned x1,
                                             unsigned* o0, unsigned* o1) {
    const unsigned ks0 = k0, ks1 = k1, ks2 = 0x1BD11BDAu ^ k0 ^ k1;
    const int ra[4] = {13, 15, 26, 6};
    const int rb[4] = {17, 29, 16, 24};
    x0 += ks0; x1 += ks1;
#pragma unroll
    for (int i = 0; i < 4; ++i) { x0 += x1; x1 = rotl32(x1, ra[i]); x1 ^= x0; }
    x0 += ks1; x1 += ks2 + 1u;
#pragma unroll
    for (int i = 0; i < 4; ++i) { x0 += x1; x1 = rotl32(x1, rb[i]); x1 ^= x0; }
    x0 += ks2; x1 += ks0 + 2u;
#pragma unroll
    for (int i = 0; i < 4; ++i) { x0 += x1; x1 = rotl32(x1, ra[i]); x1 ^= x0; }
    x0 += ks0; x1 += ks1 + 3u;
#pragma unroll
    for (int i = 0; i < 4; ++i) { x0 += x1; x1 = rotl32(x1, rb[i]); x1 ^= x0; }
    x0 += ks1; x1 += ks2 + 4u;
#pragma unroll
    for (int i = 0; i < 4; ++i) { x0 += x1; x1 = rotl32(x1, ra[i]); x1 ^= x0; }
    x0 += ks2; x1 += ks0 + 5u;
    *o0 = x0; *o1 = x1;
}

__device__ __forceinline__ float erfinv_f(float u) {
    float w = -logf((1.0f - u) * (1.0f + u));
    float p;
    if (w < 5.0f) {
        w -= 2.5f;
        p =  2.81022636e-08f;
        p =  3.43273939e-07f + p * w;
        p = -3.5233877e-06f  + p * w;
        p = -4.39150654e-06f + p * w;
        p =  0.00021858087f  + p * w;
        p = -0.00125372503f  + p * w;
        p = -0.00417768164f  + p * w;
        p =  0.246640727f    + p * w;
        p =  1.50140941f     + p * w;
    } else {
        w = sqrtf(w) - 3.0f;
        p = -0.000200214257f;
        p =  0.000100950558f + p * w;
        p =  0.00134934322f  + p * w;
        p = -0.00367342844f  + p * w;
        p =  0.00573950773f  + p * w;
        p = -0.0076224613f   + p * w;
        p =  0.00943887047f  + p * w;
        p =  1.00167406f     + p * w;
        p =  2.83297682f     + p * w;
    }
    return p * u;
}

// ---------------------------------------------------------------------------
// Kernel 2: per-batch parameter stage.
//  - threefry noise (32x5, key(42)) reproducing jax.random.normal
//  - gate xt, build per-(b,r) complex lerp/bias/scale params (with shifts)
//  - wr = softmax(siglog((wrank+noise)/extras[0]))
//  - zero the gram accumulator
// ---------------------------------------------------------------------------
__global__ __launch_bounds__(256) void k2_params(
    const float* __restrict__ xt,       // [32][106]
    const float* __restrict__ extras,   // [1]
    float* __restrict__ prm,            // [32][4][12]
    float* __restrict__ wr,             // [32][4]
    float* __restrict__ gram)           // [32][6]
{
    __shared__ float snoise[160];
    const int tid = threadIdx.x;

    if (tid < 80) {
        unsigned o0, o1;
        threefry2x32(0u, 42u, (unsigned)tid, (unsigned)(80 + tid), &o0, &o1);
        const float lo = -0.99999994f;  // nextafter(-1,0) in f32
        unsigned bits[2] = {o0, o1};
#pragma unroll
        for (int h = 0; h < 2; ++h) {
            float f01 = __uint_as_float((bits[h] >> 9) | 0x3f800000u) - 1.0f;
            float u = fmaxf(f01 * (1.0f - lo) + lo, lo);
            snoise[tid + 80 * h] = 1.41421356237f * erfinv_f(u);
        }
    }
    if (tid < 192) gram[tid] = 0.0f;   // 32*6
    __syncthreads();

    if (tid < B_) {
        const int b = tid;
        const float* row = xt + b * OUTF_;
        // gated features xa[j] = xt[j] * siglog(xt[53+j]), j < 53
        float xa[SLICE_ATT_];
#pragma unroll 1
        for (int j = 0; j < SLICE_ATT_; ++j)
            xa[j] = row[j] * siglog_f(row[SLICE_ATT_ + j]);

        // params: mod[b][r][i][t][c] -> xa[r*12 + i*4 + t*2 + c]
#pragma unroll
        for (int r = 0; r < RANK_; ++r) {
            const int jb = r * 12;
#pragma unroll
            for (int t = 0; t < 2; ++t) {
                float* o = prm + (b * RANK_ + r) * 12 + t * 6;
                o[0] = xa[jb + 0 + t * 2 + 0] + 0.5f;  // lerp.re (+shift)
                o[1] = xa[jb + 0 + t * 2 + 1];         // lerp.im
                o[2] = xa[jb + 4 + t * 2 + 0];         // bias.re
                o[3] = xa[jb + 4 + t * 2 + 1];         // bias.im
                o[4] = xa[jb + 8 + t * 2 + 0] + 1.0f;  // scale.re (+shift)
                o[5] = xa[jb + 8 + t * 2 + 1];         // scale.im
            }
        }

        // wrank -> wr softmax
        float wrank[5];
#pragma unroll
        for (int k = 0; k < 5; ++k) wrank[k] = xa[48 + k];
        wrank[4] += 0.5f;  // 1/sqrt(RANK)
        float mean = 0.f;
#pragma unroll
        for (int k = 0; k < 5; ++k) mean += wrank[k];
        mean *= 0.2f;
        float var = 0.f;
#pragma unroll
        for (int k = 0; k < 5; ++k) { float d = wrank[k] - mean; var += d * d; }
        float std = sqrtf(var * 0.25f) + EPS_;  // ddof=1
        const float inv_e = 1.0f / extras[0];
        float t5[5], mx = -1e30f;
#pragma unroll
        for (int k = 0; k < 5; ++k) {
            float lg = wrank[k] + snoise[b * 5 + k] * std * 0.01f;
            t5[k] = siglog_f(lg * inv_e);
            mx = fmaxf(mx, t5[k]);
        }
        float den = 0.f;
#pragma unroll
        for (int k = 0; k < 5; ++k) { t5[k] = expf(t5[k] - mx); den += t5[k]; }
        float inv_den = 1.0f / den;
#pragma unroll
        for (int r = 0; r < RANK_; ++r) wr[b * RANK_ + r] = t5[r] * inv_den;
        // t5[4] weights the zero "drain" plane -> never needed.
    }
}

// ---------------------------------------------------------------------------
// wf[b, r, n..n+1]: two adjacent pixels per call so every weight access is a
// 16-byte (b128) load: planes are [.., n, n+1, ..] complex-interleaved.
// ---------------------------------------------------------------------------
__device__ __forceinline__ void compute_wf4x2(
    const float* __restrict__ Wt, const float* __restrict__ prm_b, int n /*even*/,
    float out[2][RANK_])
{
    const int n2 = 2 * n;
#pragma unroll
    for (int r = 0; r < RANK_; ++r) {
        const float* p = prm_b + r * 12;
        float wv[2][2][2];   // [pixel][t][re/im]
#pragma unroll
        for (int t = 0; t < 2; ++t) {
            const float* pt = p + t * 6;
            const int base = (r * 4 + t) * PLANE_ + n2;   // l=0 plane
            float4 z1 = *reinterpret_cast<const float4*>(Wt + base);               // l=0
            float4 z2 = *reinterpret_cast<const float4*>(Wt + base + 2 * PLANE_);  // l=1
            const float cr = pt[0], ci = pt[1];
            const float br = pt[2], bi = pt[3], sr = pt[4], si = pt[5];
            const float z1r[2] = {z1.x, z1.z}, z1i[2] = {z1.y, z1.w};
            const float z2r[2] = {z2.x, z2.z}, z2i[2] = {z2.y, z2.w};
#pragma unroll
            for (int j = 0; j < 2; ++j) {
                float wre = (1.0f - cr) * z1r[j] + cr * z2r[j] - ci * (z2i[j] - z1i[j]);
                float wim = (1.0f - cr) * z1i[j] + cr * z2i[j] + ci * (z2r[j] - z1r[j]);
                wv[j][t][0] = br + (wre * sr - wim * si);
                wv[j][t][1] = bi + (wre * si + wim * sr);
            }
        }
#pragma unroll
        for (int j = 0; j < 2; ++j) {
            float denom = sqrtf(wv[j][1][0] * wv[j][1][0] + wv[j][1][1] * wv[j][1][1] + EPS_);
            float wsum  = (wv[j][0][0] * wv[j][1][0] + wv[j][0][1] * wv[j][1][1]) / denom;
            float sg = 1.0f / (1.0f + expf(-wsum));     // stabilizer = silu + leak
            out[j][r] = wsum * sg + 0.001f * wsum;
        }
    }
}

#define K3_ITER  8
#define K3_CHUNK (256 * K3_ITER * 2)   // 4096 pixels per block

// ---------------------------------------------------------------------------
// Kernel 3: accumulate the 6 off-diagonal Gram sums per batch.
// ---------------------------------------------------------------------------
__global__ __launch_bounds__(256) void k3_gram(
    const float* __restrict__ Wt,
    const float* __restrict__ prm,
    float* __restrict__ gram)
{
    const int b = blockIdx.y;
    __shared__ float sprm[48];
    if (threadIdx.x < 48) sprm[threadIdx.x] = prm[b * 48 + threadIdx.x];
    __syncthreads();

    float acc[6] = {0.f, 0.f, 0.f, 0.f, 0.f, 0.f};
    const int n0 = blockIdx.x * K3_CHUNK + threadIdx.x * 2;
#pragma unroll 1
    for (int it = 0; it < K3_ITER; ++it) {
        const int n = n0 + it * 512;
        __builtin_prefetch(Wt + 2 * (n + 1024), 0, 1);   // global_prefetch_b8
        float wf[2][RANK_];
        compute_wf4x2(Wt, sprm, n, wf);
#pragma unroll
        for (int j = 0; j < 2; ++j) {
            acc[0] += wf[j][0] * wf[j][1]; acc[1] += wf[j][0] * wf[j][2];
            acc[2] += wf[j][0] * wf[j][3]; acc[3] += wf[j][1] * wf[j][2];
            acc[4] += wf[j][1] * wf[j][3]; acc[5] += wf[j][2] * wf[j][3];
        }
    }

    // wave32 reduction
#pragma unroll
    for (int p = 0; p < 6; ++p)
#pragma unroll
        for (int off = 16; off > 0; off >>= 1)
            acc[p] += __shfl_down(acc[p], off, 32);

    __shared__ float part[8][6];
    const int wave = threadIdx.x >> 5, lane = threadIdx.x & 31;
    if (lane == 0) {
#pragma unroll
        for (int p = 0; p < 6; ++p) part[wave][p] = acc[p];
    }
    __syncthreads();
    if (threadIdx.x == 0) {
#pragma unroll
        for (int p = 0; p < 6; ++p) {
            float s = 0.f;
#pragma unroll
            for (int w = 0; w < 8; ++w) s += part[w][p];
            atomicAdd(&gram[b * 6 + p], s);
        }
    }
}

// ---------------------------------------------------------------------------
// Kernel 4: recompute wf, apply repulsion + tiny-clamp, weight by wr, write out.
// combined coefficient: (0.3/sqrt(4)) / (RANK-1) = 0.05
// ---------------------------------------------------------------------------
__global__ __launch_bounds__(256) void k4_out(
    const float* __restrict__ Wt,
    const float* __restrict__ prm,
    const float* __restrict__ wr,
    const float* __restrict__ gram,
    float* __restrict__ out)
{
    const int b = blockIdx.y;
    __shared__ float sprm[48];
    __shared__ float scoef[16];
    __shared__ float swr[4];
    if (threadIdx.x < 48) sprm[threadIdx.x] = prm[b * 48 + threadIdx.x];
    if (threadIdx.x < 4)  swr[threadIdx.x]  = wr[b * RANK_ + threadIdx.x];
    if (threadIdx.x < 16) {
        const int r = threadIdx.x >> 2, s = threadIdx.x & 3;
        float v = 0.f;
        if (r != s) {
            const int lo = r < s ? r : s, hi = r < s ? s : r;
            const int base = (lo == 0) ? 0 : (lo == 1 ? 3 : 5);
            v = 0.05f * gram[b * 6 + base + (hi - lo - 1)];
        }
        scoef[threadIdx.x] = v;
    }
    __syncthreads();

    const int n0 = blockIdx.x * K3_CHUNK + threadIdx.x * 2;
#pragma unroll 1
    for (int it = 0; it < K3_ITER; ++it) {
        const int n = n0 + it * 512;
        __builtin_prefetch(Wt + 2 * (n + 1024), 0, 1);
        float wf[2][RANK_];
        compute_wf4x2(Wt, sprm, n, wf);
        float2 o2;
#pragma unroll
        for (int j = 0; j < 2; ++j) {
            float o = 0.f;
#pragma unroll
            for (int r = 0; r < RANK_; ++r) {
                float rep = scoef[r * 4 + 0] * wf[j][0] + scoef[r * 4 + 1] * wf[j][1]
                          + scoef[r * 4 + 2] * wf[j][2] + scoef[r * 4 + 3] * wf[j][3];
                float v = wf[j][r] - rep;
                if (fabsf(v) < EPS_)
                    v = (v > 0.f) ? EPS_ : (v < 0.f ? -EPS_ : 0.f);
                o += swr[r] * v;
            }
            if (j == 0) o2.x = o; else o2.y = o;
        }
        *reinterpret_cast<float2*>(out + b * HW_ + n) = o2;   // coalesced b64 store
    }
}

// ---------------------------------------------------------------------------
extern "C" void kernel_launch(void* const* d_in, const int* in_sizes, int n_in,
                              void* d_out, int out_size, void* d_ws, size_t ws_size,
                              hipStream_t stream) {
    const float* x      = (const float*)d_in[0];   // [32][2048]
    const float* ctx_w  = (const float*)d_in[1];   // [106][2048]
    const float* ctx_b  = (const float*)d_in[2];   // [106]
    const float* wts    = (const float*)d_in[3];   // [1,4,2,2,256,256,2]
    const float* extras = (const float*)d_in[4];   // [1]
    float* out = (float*)d_out;                    // [32][256][256]

    float* ws   = (float*)d_ws;
    float* xt   = ws;          // 32*106 = 3392 floats
    float* prm  = ws + 4096;   // 32*48  = 1536 floats
    float* wr   = ws + 8192;   // 32*4   = 128 floats
    float* gram = ws + 8448;   // 32*6   = 192 floats

    k1_gemm_wmma<<<dim3(1), dim3(448), 0, stream>>>(x, ctx_w, ctx_b, xt);
    k2_params<<<dim3(1), dim3(256), 0, stream>>>(xt, extras, prm, wr, gram);
    k3_gram<<<dim3(HW_ / K3_CHUNK, B_), dim3(256), 0, stream>>>(wts, prm, gram);
    k4_out <<<dim3(HW_ / K3_CHUNK, B_), dim3(256), 0, stream>>>(wts, prm, wr, gram, out);
}